// TickDataTokenizer_40355512714014
// MI455X (gfx1250) — compile-verified
//
#include <hip/hip_runtime.h>
#include <math.h>

typedef __attribute__((ext_vector_type(2))) float v2f;
typedef __attribute__((ext_vector_type(8))) float v8f;

#define T_LEN  100000
#define PATCH  50
#define NFEAT  7
#define DH     128
#define AGG    512             // 4*DH
#define DMODEL 256
#define DQ     64
#define NPATCH (T_LEN / PATCH) // 2000
#define COMB   (DMODEL + DQ)   // 320
#define LN_EPS 1e-5f

// ---------------------------------------------------------------------------
// Kernel 1: one block per patch.
//  - per-channel tick projection with fused mean/std/min/max stats (registers)
//  - LN1 over the 512-wide agg row  -> anorm workspace
//  - micro features + micro embedding -> comb[:, 256:320]
// ---------------------------------------------------------------------------
__global__ __launch_bounds__(128) void k_patch(
        const float* __restrict__ ticks,
        const float* __restrict__ tpw,
        const float* __restrict__ ln1g, const float* __restrict__ ln1b,
        const float* __restrict__ microw,
        float* __restrict__ anorm, float* __restrict__ comb)
{
    __shared__ float s_ticks[PATCH * NFEAT];   // 350
    __shared__ float s_w[DH * NFEAT];          // 896
    __shared__ float s_agg[AGG];
    __shared__ float s_micro[8];
    __shared__ float s_red[128], s_red2[128];
    __shared__ float s_stats[2];

    const int p   = blockIdx.x;                // 0 .. B*NPATCH-1
    const int b   = p / NPATCH;
    const int n   = p % NPATCH;
    const int tid = threadIdx.x;

    const float* tick_base = ticks + ((size_t)b * T_LEN + (size_t)n * PATCH) * NFEAT;
    for (int i = tid; i < PATCH * NFEAT; i += 128) s_ticks[i] = tick_base[i];
    for (int i = tid; i < DH * NFEAT;    i += 128) s_w[i]     = tpw[i];
    __syncthreads();

    // ---- per-channel projection + patch statistics (channel d = tid) ----
    {
        const int d = tid;
        const float w0 = s_w[d*7+0], w1 = s_w[d*7+1], w2 = s_w[d*7+2],
                    w3 = s_w[d*7+3], w4 = s_w[d*7+4], w5 = s_w[d*7+5],
                    w6 = s_w[d*7+6];
        float sum = 0.f, sq = 0.f, mn = 1e30f, mx = -1e30f;
        for (int t = 0; t < PATCH; ++t) {
            const float* x = &s_ticks[t*7];
            float e = x[0]*w0 + x[1]*w1 + x[2]*w2 + x[3]*w3
                    + x[4]*w4 + x[5]*w5 + x[6]*w6;
            sum += e; sq += e*e;
            mn = fminf(mn, e); mx = fmaxf(mx, e);
        }
        float mean = sum * (1.f / PATCH);
        float varn = sq - (float)PATCH * mean * mean;     // Σ(x-m)^2
        float stdv = sqrtf(fmaxf(varn, 0.f) * (1.f / (PATCH - 1)));
        s_agg[d]        = mean;
        s_agg[DH + d]   = stdv;
        s_agg[2*DH + d] = mn;
        s_agg[3*DH + d] = mx;
    }

    // ---- microstructure features (thread 0, 50 serial iterations) ----
    if (tid == 0) {
        float psum = 0.f, psq = 0.f, vsum = 0.f, vmax = -1e30f;
        float buy = 0.f, sell = 0.f, ssum = 0.f;
        float pmin = 1e30f, pmax = -1e30f, pfirst = 0.f, plast = 0.f;
        for (int t = 0; t < PATCH; ++t) {
            float pr = s_ticks[t*7+0];
            float vo = s_ticks[t*7+1];
            float sd = s_ticks[t*7+2];
            psum += pr; psq += pr*pr;
            vsum += vo; vmax = fmaxf(vmax, vo);
            if (sd > 0.f) buy  += vo;
            if (sd < 0.f) sell += vo;
            ssum += sd;
            pmin = fminf(pmin, pr); pmax = fmaxf(pmax, pr);
            if (t == 0)         pfirst = pr;
            if (t == PATCH - 1) plast  = pr;
        }
        float pmean = psum * (1.f / PATCH);
        float pvarn = psq - (float)PATCH * pmean * pmean;
        s_micro[0] = pmean;
        s_micro[1] = sqrtf(fmaxf(pvarn, 0.f) * (1.f / (PATCH - 1)));
        s_micro[2] = log1pf(vsum);
        s_micro[3] = (buy - sell) / fmaxf(vsum, 1e-8f);
        s_micro[4] = plast - pfirst;
        s_micro[5] = ssum * (1.f / PATCH);
        s_micro[6] = pmax - pmin;
        s_micro[7] = log1pf(vmax);
    }
    __syncthreads();

    // ---- micro embedding -> comb[p][256 + q] ----
    if (tid < DQ) {
        float acc = 0.f;
        #pragma unroll
        for (int j = 0; j < 8; ++j) acc += s_micro[j] * microw[tid*8 + j];
        comb[(size_t)p * COMB + DMODEL + tid] = acc;
    }

    // ---- LN1 over s_agg[512] -> anorm ----
    {
        float s = 0.f, q = 0.f;
        #pragma unroll
        for (int j = 0; j < 4; ++j) { float x = s_agg[tid*4 + j]; s += x; q += x*x; }
        s_red[tid] = s; s_red2[tid] = q;
        __syncthreads();
        for (int off = 64; off > 0; off >>= 1) {
            if (tid < off) { s_red[tid] += s_red[tid+off]; s_red2[tid] += s_red2[tid+off]; }
            __syncthreads();
        }
        if (tid == 0) {
            float mean = s_red[0] * (1.f / AGG);
            float var  = s_red2[0] * (1.f / AGG) - mean * mean;
            s_stats[0] = mean;
            s_stats[1] = rsqrtf(var + LN_EPS);
        }
        __syncthreads();
        float mean = s_stats[0], inv = s_stats[1];
        float* arow = anorm + (size_t)p * AGG;
        #pragma unroll
        for (int j = 0; j < 4; ++j) {
            int i = tid*4 + j;
            arow[i] = (s_agg[i] - mean) * inv * ln1g[i] + ln1b[i];
        }
    }
}

// ---------------------------------------------------------------------------
// Kernel 2: patch_emb = silu(anorm @ patch_w^T + patch_b) -> comb[:, 0:256]
// WMMA f32 16x16x4.  Block = 8 waves; M-tile 32 (two 16-row subtiles reuse
// each B fragment -> 4 WMMAs per 4 fragment loads); N covered fully.
// ---------------------------------------------------------------------------
__global__ __launch_bounds__(256) void k_gemm1(
        const float* __restrict__ anorm,
        const float* __restrict__ pw, const float* __restrict__ pb,
        float* __restrict__ comb)
{
    __shared__ float s_a[32 * AGG];            // 64 KB A-tile
    const int tid = threadIdx.x;
    const int m0  = blockIdx.x * 32;

    {   // vectorized stage: 32 rows x 512 cols, contiguous
        const float4* src = (const float4*)(anorm + (size_t)m0 * AGG);
        float4*       dst = (float4*)s_a;
        for (int i = tid; i < 32 * AGG / 4; i += 256) dst[i] = src[i];
    }
    __syncthreads();

    const int wave = tid >> 5;
    const int lane = tid & 31;
    const int lr   = lane & 15;    // M (for A) / N (for B) within tile
    const int lh   = lane >> 4;    // selects K pair {0,1} vs {2,3}
    const int n0   = wave * 32;

    v8f acc00 = {}, acc01 = {}, acc10 = {}, acc11 = {};

    const float* a0row = &s_a[lr * AGG];
    const float* a1row = &s_a[(16 + lr) * AGG];
    const float* b0row = pw + (size_t)(n0 +      lr) * AGG;
    const float* b1row = pw + (size_t)(n0 + 16 + lr) * AGG;

    for (int k = 0; k < AGG; k += 4) {
        const int ko = k + 2*lh;
        v2f a0 = *(const v2f*)(a0row + ko);    // ds_load_b64
        v2f a1 = *(const v2f*)(a1row + ko);
        v2f b0 = *(const v2f*)(b0row + ko);    // global_load_b64
        v2f b1 = *(const v2f*)(b1row + ko);
        acc00 = __builtin_amdgcn_wmma_f32_16x16x4_f32(false, a0, false, b0,
                                                      (short)0, acc00, false, false);
        acc01 = __builtin_amdgcn_wmma_f32_16x16x4_f32(false, a0, false, b1,
                                                      (short)0, acc01, false, false);
        acc10 = __builtin_amdgcn_wmma_f32_16x16x4_f32(false, a1, false, b0,
                                                      (short)0, acc10, false, false);
        acc11 = __builtin_amdgcn_wmma_f32_16x16x4_f32(false, a1, false, b1,
                                                      (short)0, acc11, false, false);
    }

    // C layout: N = lane%16, M = r + 8*(lane/16)
    const float bias0 = pb[n0 + lr];
    const float bias1 = pb[n0 + 16 + lr];
    #pragma unroll
    for (int r = 0; r < 8; ++r) {
        const int mA = m0 +      r + 8*lh;
        const int mB = m0 + 16 + r + 8*lh;
        float v;
        v = acc00[r] + bias0;
        comb[(size_t)mA * COMB + n0 +      lr] = v * __builtin_amdgcn_rcpf(1.f + __expf(-v));
        v = acc01[r] + bias1;
        comb[(size_t)mA * COMB + n0 + 16 + lr] = v * __builtin_amdgcn_rcpf(1.f + __expf(-v));
        v = acc10[r] + bias0;
        comb[(size_t)mB * COMB + n0 +      lr] = v * __builtin_amdgcn_rcpf(1.f + __expf(-v));
        v = acc11[r] + bias1;
        comb[(size_t)mB * COMB + n0 + 16 + lr] = v * __builtin_amdgcn_rcpf(1.f + __expf(-v));
    }
}

// ---------------------------------------------------------------------------
// Kernel 3: out = LN2(comb @ final_w^T).  Same M-tile-32 WMMA tiling (K=320);
// C parked in LDS so the 256-wide LayerNorm is done in-block with wave32
// shuffles (wave w owns rows 4w..4w+3).
// ---------------------------------------------------------------------------
__global__ __launch_bounds__(256) void k_gemm2_ln(
        const float* __restrict__ comb,
        const float* __restrict__ fw,
        const float* __restrict__ ln2g, const float* __restrict__ ln2b,
        float* __restrict__ out)
{
    __shared__ float s_a[32 * COMB];           // 40 KB
    __shared__ float s_c[32 * DMODEL];         // 32 KB
    const int tid = threadIdx.x;
    const int m0  = blockIdx.x * 32;

    {   // vectorized stage: 32 rows x 320 cols, contiguous
        const float4* src = (const float4*)(comb + (size_t)m0 * COMB);
        float4*       dst = (float4*)s_a;
        for (int i = tid; i < 32 * COMB / 4; i += 256) dst[i] = src[i];
    }
    __syncthreads();

    const int wave = tid >> 5;
    const int lane = tid & 31;
    const int lr   = lane & 15;
    const int lh   = lane >> 4;
    const int n0   = wave * 32;

    v8f acc00 = {}, acc01 = {}, acc10 = {}, acc11 = {};

    const float* a0row = &s_a[lr * COMB];
    const float* a1row = &s_a[(16 + lr) * COMB];
    const float* b0row = fw + (size_t)(n0 +      lr) * COMB;
    const float* b1row = fw + (size_t)(n0 + 16 + lr) * COMB;

    for (int k = 0; k < COMB; k += 4) {
        const int ko = k + 2*lh;
        v2f a0 = *(const v2f*)(a0row + ko);
        v2f a1 = *(const v2f*)(a1row + ko);
        v2f b0 = *(const v2f*)(b0row + ko);
        v2f b1 = *(const v2f*)(b1row + ko);
        acc00 = __builtin_amdgcn_wmma_f32_16x16x4_f32(false, a0, false, b0,
                                                      (short)0, acc00, false, false);
        acc01 = __builtin_amdgcn_wmma_f32_16x16x4_f32(false, a0, false, b1,
                                                      (short)0, acc01, false, false);
        acc10 = __builtin_amdgcn_wmma_f32_16x16x4_f32(false, a1, false, b0,
                                                      (short)0, acc10, false, false);
        acc11 = __builtin_amdgcn_wmma_f32_16x16x4_f32(false, a1, false, b1,
                                                      (short)0, acc11, false, false);
    }

    #pragma unroll
    for (int r = 0; r < 8; ++r) {
        const int rowA = r + 8*lh;
        const int rowB = 16 + r + 8*lh;
        s_c[rowA * DMODEL + n0 +      lr] = acc00[r];
        s_c[rowA * DMODEL + n0 + 16 + lr] = acc01[r];
        s_c[rowB * DMODEL + n0 +      lr] = acc10[r];
        s_c[rowB * DMODEL + n0 + 16 + lr] = acc11[r];
    }
    __syncthreads();

    // LayerNorm over each 256-wide row; wave w owns rows 4w..4w+3.
    #pragma unroll
    for (int rr = 0; rr < 4; ++rr) {
        const int row = wave * 4 + rr;
        const float* crow = &s_c[row * DMODEL];
        float s = 0.f, q = 0.f;
        #pragma unroll
        for (int j = 0; j < 8; ++j) {
            float x = crow[lane*8 + j];
            s += x; q += x*x;
        }
        for (int off = 16; off > 0; off >>= 1) {
            s += __shfl_xor(s, off, 32);
            q += __shfl_xor(q, off, 32);
        }
        const float mean = s * (1.f / DMODEL);
        const float var  = q * (1.f / DMODEL) - mean * mean;
        const float inv  = rsqrtf(var + LN_EPS);
        float* orow = out + (size_t)(m0 + row) * DMODEL;
        #pragma unroll
        for (int j = 0; j < 8; ++j) {
            const int c = lane*8 + j;
            orow[c] = (crow[c] - mean) * inv * ln2g[c] + ln2b[c];
        }
    }
}

// ---------------------------------------------------------------------------
extern "C" void kernel_launch(void* const* d_in, const int* in_sizes, int n_in,
                              void* d_out, int out_size, void* d_ws, size_t ws_size,
                              hipStream_t stream)
{
    (void)n_in; (void)out_size; (void)ws_size;

    const float* ticks = (const float*)d_in[0];
    const float* tpw   = (const float*)d_in[1];
    const float* ln1g  = (const float*)d_in[2];
    const float* ln1b  = (const float*)d_in[3];
    const float* pw    = (const float*)d_in[4];
    const float* pb    = (const float*)d_in[5];
    const float* mw    = (const float*)d_in[6];
    const float* fw    = (const float*)d_in[7];
    const float* ln2g  = (const float*)d_in[8];
    const float* ln2b  = (const float*)d_in[9];

    const int B = in_sizes[0] / (T_LEN * NFEAT);   // 8
    const int M = B * NPATCH;                      // 16000 rows

    float* anorm = (float*)d_ws;                                            // M x 512
    float* comb  = (float*)((char*)d_ws + (size_t)M * AGG * sizeof(float)); // M x 320
    float* outp  = (float*)d_out;                                           // M x 256

    k_patch   <<<M,      128, 0, stream>>>(ticks, tpw, ln1g, ln1b, mw, anorm, comb);
    k_gemm1   <<<M / 32, 256, 0, stream>>>(anorm, pw, pb, comb);
    k_gemm2_ln<<<M / 32, 256, 0, stream>>>(comb, fw, ln2g, ln2b, outp);
}